// CruxMiniCircuit_27144193310723
// MI455X (gfx1250) — compile-verified
//
#include <hip/hip_runtime.h>
#include <stdint.h>

// ---------------- problem constants ----------------
#define BB     2048
#define NN     1023
#define NI     10
#define NOPS   3
#define NPASS  10
#define STR    12            // padded state row stride in floats (48B, 16B aligned)
#define KST    25            // k-steps of 4 -> K = 100
#define BSZ    256           // 8 wave32s
#define NWAVE  8
#define TPW    8             // node-tiles (of 16) per wave: 8*8*16 = 1024 >= 1023

// ---------------- LDS layout (bytes) ----------------
#define OFF_STA   0
#define SZ_ST     (NN*STR*4)            // 49104
#define OFF_STB   (OFF_STA + SZ_ST)     // 49104
#define OFF_BTAB  (OFF_STB + SZ_ST)     // 98208 (8B aligned)
#define SZ_BTAB   (KST*2*32*8)          // 12800 (float2 per lane per kstep per coltile)
#define OFF_SCR   (OFF_BTAB + SZ_BTAB)  // 111008
#define SZ_SCR    (NWAVE*16*32*4)       // 16384
#define OFF_LEFT  (OFF_SCR + SZ_SCR)    // 127392
#define OFF_RIGHT (OFF_LEFT  + NN*4)
#define OFF_OP    (OFF_RIGHT + NN*4)
#define OFF_ISOP  (OFF_OP    + NN*4)
#define OFF_ROOT  (OFF_ISOP  + NN*4)
#define SMEM_BYTES (OFF_ROOT + 16*4)    // 143824 < 320KB

typedef float v2f __attribute__((ext_vector_type(2)));
typedef float v4f __attribute__((ext_vector_type(4)));
typedef float v8f __attribute__((ext_vector_type(8)));

__global__ __launch_bounds__(BSZ)
void circuit_eval_kernel(const float* __restrict__ op_table,
                         const int*   __restrict__ cats,
                         const int*   __restrict__ ops,
                         const int*   __restrict__ lits,
                         const int*   __restrict__ left,
                         const int*   __restrict__ right,
                         const unsigned char* __restrict__ mask,
                         float* __restrict__ out)
{
    extern __shared__ char smem[];
    float* stA    = (float*)(smem + OFF_STA);
    float* stB    = (float*)(smem + OFF_STB);
    v2f*   btab   = (v2f*)  (smem + OFF_BTAB);
    float* scr    = (float*)(smem + OFF_SCR);
    int*   sLeft  = (int*)  (smem + OFF_LEFT);
    int*   sRight = (int*)  (smem + OFF_RIGHT);
    int*   sOp    = (int*)  (smem + OFF_OP);
    int*   sIsOp  = (int*)  (smem + OFF_ISOP);
    float* sRoot  = (float*)(smem + OFF_ROOT);

    const int  b   = blockIdx.x;
    const int  tid = threadIdx.x;
    const long g   = (long)b * NN;

    // ---------- init: metadata, initial state (lit one-hots) ----------
    for (int n = tid; n < NN; n += BSZ) {
        int cat = cats[g + n];
        int mk  = mask[g + n] ? 1 : 0;
        int lt  = lits[g + n];  lt = lt < 0 ? 0 : (lt > NI - 1   ? NI - 1   : lt);
        int li  = left[g + n];  li = li < 0 ? 0 : (li > NN - 1   ? NN - 1   : li);
        int ri  = right[g + n]; ri = ri < 0 ? 0 : (ri > NN - 1   ? NN - 1   : ri);
        int op  = ops[g + n];   op = op < 0 ? 0 : (op > NOPS - 1 ? NOPS - 1 : op);
        sLeft[n] = li; sRight[n] = ri; sOp[n] = op;
        sIsOp[n] = (cat == 1) && mk;
        int litm = (cat == 0) && mk;
        #pragma unroll
        for (int k = 0; k < STR; ++k)
            stA[n * STR + k] = (litm && k == lt) ? 1.0f : 0.0f;
    }

    // ---------- init: pre-swizzle op_table into WMMA B-fragment layout ----------
    // btab[kk*64 + tile*32 + lane] = {B[K0][col], B[K0+1][col]}
    //   lane<16 : K0 = 4kk,   col = tile*16 + lane
    //   lane>=16: K0 = 4kk+2, col = tile*16 + lane-16
    //   B[K][col] = op_table[col/10][K/10][K%10][col%10], cols 30/31 = 0 pad
    for (int e = tid; e < KST * 2 * 32; e += BSZ) {
        int kk   = e >> 6;
        int r    = e & 63;
        int tile = r >> 5;
        int lane = r & 31;
        int K0   = 4 * kk + ((lane >= 16) ? 2 : 0);
        int col  = tile * 16 + (lane & 15);
        v2f v; v.x = 0.0f; v.y = 0.0f;
        if (col < NOPS * NI) {
            int o = col / NI, ko = col % NI;
            int i0 = K0 / NI,      j0 = K0 % NI;
            int i1 = (K0 + 1) / NI, j1 = (K0 + 1) % NI;
            v.x = op_table[((o * NI + i0) * NI + j0) * NI + ko];
            v.y = op_table[((o * NI + i1) * NI + j1) * NI + ko];
        }
        btab[e] = v;
    }
    __syncthreads();

    float* stOld = stA;
    float* stNew = stB;
    const int  lane   = tid & 31;
    const int  wv     = tid >> 5;
    const int  lane16 = lane & 15;
    const bool hiHalf = lane >= 16;
    float* sc = scr + wv * (16 * 32);

    for (int pass = 0; pass < NPASS; ++pass) {
        const bool last = (pass == NPASS - 1);
        for (int t = 0; t < TPW; ++t) {
            const int  tileBase = (wv * TPW + t) * 16;
            const int  m     = tileBase + lane16;
            const bool valid = m < NN;
            const int  mm    = valid ? m : 0;
            const int  li = sLeft[mm], ri = sRight[mm];

            // gather children's distributions (LDS, vectorized: 48B padded rows)
            const float* Lrow = stOld + li * STR;
            const float* Rrow = stOld + ri * STR;
            v4f L0 = *(const v4f*)(Lrow);     v4f L1 = *(const v4f*)(Lrow + 4);
            v2f L2 = *(const v2f*)(Lrow + 8);
            v4f R0 = *(const v4f*)(Rrow);     v4f R1 = *(const v4f*)(Rrow + 4);
            v2f R2 = *(const v2f*)(Rrow + 8);
            float Lr[10] = {L0.x,L0.y,L0.z,L0.w,L1.x,L1.y,L1.z,L1.w,L2.x,L2.y};
            float Rr[10] = {R0.x,R0.y,R0.z,R0.w,R1.x,R1.y,R1.z,R1.w,R2.x,R2.y};

            // (16 nodes x 100) @ (100 x 32) via f32 WMMA, K in steps of 4
            v8f c0 = {0,0,0,0,0,0,0,0};
            v8f c1 = {0,0,0,0,0,0,0,0};
            #pragma unroll
            for (int kk = 0; kk < KST; ++kk) {
                const int K = 4 * kk;
                // A fragment: lanes 0-15 hold K,K+1 ; lanes 16-31 hold K+2,K+3
                float pL0 = Lr[(K + 0) / NI] * Rr[(K + 0) % NI];
                float pL1 = Lr[(K + 1) / NI] * Rr[(K + 1) % NI];
                float pH0 = Lr[(K + 2) / NI] * Rr[(K + 2) % NI];
                float pH1 = Lr[(K + 3) / NI] * Rr[(K + 3) % NI];
                v2f a;
                a.x = hiHalf ? pH0 : pL0;
                a.y = hiHalf ? pH1 : pL1;
                v2f b0 = btab[kk * 64 + lane];        // col tile 0 (cols 0-15)
                v2f b1 = btab[kk * 64 + 32 + lane];   // col tile 1 (cols 16-31)
                c0 = __builtin_amdgcn_wmma_f32_16x16x4_f32(false, a, false, b0,
                                                           (short)0, c0, false, false);
                c1 = __builtin_amdgcn_wmma_f32_16x16x4_f32(false, a, false, b1,
                                                           (short)0, c1, false, false);
            }

            // dump D tiles to wave-private scratch: scr[M][col]
            const int Mb = hiHalf ? 8 : 0;
            #pragma unroll
            for (int v = 0; v < 8; ++v) {
                sc[(Mb + v) * 32 + lane16]      = c0[v];
                sc[(Mb + v) * 32 + 16 + lane16] = c1[v];
            }
            asm volatile("s_wait_dscnt 0" ::: "memory");  // lockstep wave: stores visible

            // per-node softmax + masked state update (lanes 0-15, one node each)
            if (!hiHalf && valid) {
                const int o = sOp[m];
                const float* lrow = sc + lane16 * 32 + o * NI;
                float lg[NI];
                float mx = -3.0e38f;
                #pragma unroll
                for (int k = 0; k < NI; ++k) { lg[k] = lrow[k]; mx = fmaxf(mx, lg[k]); }
                if (last && wv == 0 && t == 0 && m == 0) {
                    #pragma unroll
                    for (int k = 0; k < NI; ++k) sRoot[k] = lg[k];  // root node_logits
                }
                float s = 0.0f;
                #pragma unroll
                for (int k = 0; k < NI; ++k) { lg[k] = __expf(lg[k] - mx); s += lg[k]; }
                const float inv = 1.0f / s;
                const int isop  = sIsOp[m];
                float*       dst = stNew + m * STR;
                const float* src = stOld + m * STR;
                #pragma unroll
                for (int k = 0; k < NI; ++k)
                    dst[k] = isop ? lg[k] * inv : src[k];
            }
            asm volatile("s_wait_dscnt 0" ::: "memory");  // scratch reuse next tile
        }
        __syncthreads();
        float* tmp = stOld; stOld = stNew; stNew = tmp;
    }

    // root output: lit -> 10*state, else last-pass logits
    if (tid < NI) {
        const int cat0 = cats[g];
        const float v = (cat0 == 0) ? stOld[tid] * 10.0f : sRoot[tid];
        out[(long)b * NI + tid] = v;
    }
}

extern "C" void kernel_launch(void* const* d_in, const int* in_sizes, int n_in,
                              void* d_out, int out_size, void* d_ws, size_t ws_size,
                              hipStream_t stream) {
    (void)in_sizes; (void)n_in; (void)out_size; (void)d_ws; (void)ws_size;
    const float*         op_table = (const float*)d_in[0];
    const int*           cats     = (const int*)d_in[1];
    const int*           ops      = (const int*)d_in[2];
    const int*           lits     = (const int*)d_in[3];
    const int*           left     = (const int*)d_in[4];
    const int*           right    = (const int*)d_in[5];
    const unsigned char* mask     = (const unsigned char*)d_in[6];
    float* out = (float*)d_out;

    static int attr_set = 0;
    if (!attr_set) {
        hipFuncSetAttribute((const void*)circuit_eval_kernel,
                            hipFuncAttributeMaxDynamicSharedMemorySize, SMEM_BYTES);
        attr_set = 1;
    }
    circuit_eval_kernel<<<BB, BSZ, SMEM_BYTES, stream>>>(op_table, cats, ops, lits,
                                                         left, right, mask, out);
}